// EGATLayer_70153995813489
// MI455X (gfx1250) — compile-verified
//
#include <hip/hip_runtime.h>
#include <hip/hip_bf16.h>
#include <math.h>

// EGAT layer for MI455X (gfx1250). fp32 end-to-end using V_WMMA_F32_16X16X4_F32,
// with Tensor-Data-Mover (tensor_load_to_lds) staging of GEMM A-tiles.
// H=4 heads, D=32, H*D=128; N=50000 nodes, E=600000 edges, IN=128, ED=64.

#define HH 4
#define DD 32
#define HD 128
#define NEG_SLOPE 0.01f
#define ROWS 64   // output rows per GEMM block (4 WMMA M-tiles per wave)

typedef __attribute__((ext_vector_type(2))) float v2f;
typedef __attribute__((ext_vector_type(4))) float v4f;
typedef __attribute__((ext_vector_type(8))) float v8f;
typedef __attribute__((ext_vector_type(4))) unsigned int u32x4;
typedef __attribute__((ext_vector_type(4))) int i32x4;
typedef __attribute__((ext_vector_type(8))) int i32x8;

// ---------------------------------------------------------------------------
// TDM: DMA a [rows_rem x K] fp32 2D tile (ROWS rows wanted, OOB rows zero-filled)
// from global into LDS with a padded row stride of K+4 dwords.
// Descriptor layout per CDNA5 ISA 8.3/8.4 (D# group0 = 4 SGPRs, group1 = 8).
// ---------------------------------------------------------------------------
#if __has_builtin(__builtin_amdgcn_tensor_load_to_lds)
#define HAVE_TDM 1
__device__ inline void tdm_load_tile_f32(const float* gptr, unsigned lds_byte_off,
                                         int rows_rem, int K)
{
    unsigned long long ga = (unsigned long long)(uintptr_t)gptr;
    u32x4 g0;
    g0[0] = 1u;                                        // count=1, user mode, no gather
    g0[1] = lds_byte_off;                              // lds_addr [63:32]
    g0[2] = (unsigned)ga;                              // global_addr [95:64]
    g0[3] = (unsigned)((ga >> 32) & 0x01ffffffu)       // global_addr [120:96]
          | (2u << 30);                                // type=2 ("image") [127:126]
    const int pad_int = (K == 128) ? 6 : 5;            // pad after 128 / 64 dwords
    i32x8 g1;
    g1[0] = (2 << 16)                                  // data_size = 4 bytes
          | (1 << 20)                                  // pad_enable
          | (pad_int << 22)                            // pad_interval
          | (3 << 25);                                 // pad_amount = 4 dwords
    g1[1] = (K & 0xffff) << 16;                        // tensor_dim0[15:0]
    g1[2] = ((unsigned)K >> 16) | ((rows_rem & 0xffff) << 16);   // dim0 hi | dim1 lo
    g1[3] = (((unsigned)rows_rem >> 16) & 0xffff) | (K << 16);   // dim1 hi | tile_dim0
    g1[4] = ROWS;                                      // tile_dim1 = 64, tile_dim2 = 0
    g1[5] = K;                                         // tensor_dim0_stride [31:0]
    g1[6] = 0;                                         // stride hi | dim1_stride lo
    g1[7] = 0;
    i32x4 z4 = {0, 0, 0, 0};
#if defined(__clang_major__) && (__clang_major__ >= 23)
    i32x8 z8 = {0, 0, 0, 0, 0, 0, 0, 0};
    __builtin_amdgcn_tensor_load_to_lds(g0, g1, z4, z4, z8, 0);
#else
    __builtin_amdgcn_tensor_load_to_lds(g0, g1, z4, z4, 0);
#endif
    __builtin_amdgcn_s_wait_tensorcnt(0);
}
#else
#define HAVE_TDM 0
#endif

// ---------------------------------------------------------------------------
// fp32 GEMM: C[M,128] = A[M,K] @ B[K,128], K in {64,128}.
// Block = 64 output rows, 8 waves each own a 16-col tile and 4 M-tiles,
// so each B fragment feeds 4 independent v_wmma_f32_16x16x4_f32.
// A tile staged in LDS by the TDM (padded stride K+4 -> conflict-free
// ds_load_2addr_b32 fragments). WMMA f32 layouts per ISA 7.12.2.
// ---------------------------------------------------------------------------
__global__ __launch_bounds__(256) void gemm_f32_wmma(
    const float* __restrict__ A, const float* __restrict__ B,
    float* __restrict__ C, int M, int K, int Ncols)
{
    __shared__ __align__(16) float sA[ROWS * 132];     // supports K up to 128
    const int strideA = K + 4;
    const int tileM = blockIdx.x * ROWS;
    const int tid = threadIdx.x;

#if HAVE_TDM
    if ((tid >> 5) == 0) {   // wave 0 issues the DMA and waits TENSORcnt
        tdm_load_tile_f32(A + (size_t)tileM * K,
                          (unsigned)(uintptr_t)(&sA[0]), M - tileM, K);
    }
#else
    const int K4 = K >> 2;
    for (int i = tid; i < ROWS * K4; i += 256) {
        int r = i / K4, c4 = i - r * K4;
        int row = tileM + r; if (row >= M) row = M - 1;
        v4f v = *reinterpret_cast<const v4f*>(A + (size_t)row * K + c4 * 4);
        *reinterpret_cast<v4f*>(&sA[r * strideA + c4 * 4]) = v;
    }
#endif
    __syncthreads();

    const int wave = tid >> 5, lane = tid & 31;
    const int lo = lane & 15, hi = lane >> 4;          // hi selects K pair {0,1} vs {2,3}
    const int col0 = wave << 4;                        // Ncols == 128 -> 8 col tiles
    v8f acc0 = {0.f, 0.f, 0.f, 0.f, 0.f, 0.f, 0.f, 0.f};
    v8f acc1 = acc0, acc2 = acc0, acc3 = acc0;

    for (int k0 = 0; k0 < K; k0 += 4) {
        const int ka = k0 + 2 * hi;
        v2f b;
        b.x = B[(size_t)ka * Ncols + col0 + lo];
        b.y = B[(size_t)(ka + 1) * Ncols + col0 + lo];
        v2f a0, a1, a2, a3;
        const int base = lo * strideA + ka;
        a0.x = sA[base + 0 * 16 * strideA]; a0.y = sA[base + 0 * 16 * strideA + 1];
        a1.x = sA[base + 1 * 16 * strideA]; a1.y = sA[base + 1 * 16 * strideA + 1];
        a2.x = sA[base + 2 * 16 * strideA]; a2.y = sA[base + 2 * 16 * strideA + 1];
        a3.x = sA[base + 3 * 16 * strideA]; a3.y = sA[base + 3 * 16 * strideA + 1];
        acc0 = __builtin_amdgcn_wmma_f32_16x16x4_f32(false, a0, false, b, (short)0, acc0, false, false);
        acc1 = __builtin_amdgcn_wmma_f32_16x16x4_f32(false, a1, false, b, (short)0, acc1, false, false);
        acc2 = __builtin_amdgcn_wmma_f32_16x16x4_f32(false, a2, false, b, (short)0, acc2, false, false);
        acc3 = __builtin_amdgcn_wmma_f32_16x16x4_f32(false, a3, false, b, (short)0, acc3, false, false);
    }

    // C/D layout: vgpr j -> row j + 8*hi, col = lane%16 (ISA 7.12.2)
    v8f accs[4] = {acc0, acc1, acc2, acc3};
    for (int t = 0; t < 4; ++t) {
        for (int j = 0; j < 8; ++j) {
            int row = tileM + t * 16 + j + 8 * hi;
            if (row < M) C[(size_t)row * Ncols + col0 + lo] = accs[t][j];
        }
    }
}

// ---------------------------------------------------------------------------
// init: emax = -inf, denom = 0, agg = 0
// ---------------------------------------------------------------------------
__global__ void init_kernel(float* __restrict__ emax, float* __restrict__ denom,
                            float* __restrict__ agg, int NH, int NHD)
{
    int i = blockIdx.x * blockDim.x + threadIdx.x;
    if (i < NH) { emax[i] = -__builtin_inff(); denom[i] = 0.f; }
    if (i < NHD) agg[i] = 0.f;
}

// sign-aware float atomic max: maps to native global_atomic_max_i32 / min_u32
__device__ inline void atomicMaxF(float* addr, float val)
{
    if (val >= 0.f) atomicMax((int*)addr, __float_as_int(val));
    else            atomicMin((unsigned int*)addr, __float_as_uint(val));
}

// ---------------------------------------------------------------------------
// edge_logits: one wave per edge. lanes cover D=32; loop over H=4 heads.
// e = leaky_relu( dot(zn[src]+ze, Wa[:32]) + dot(zn[dst]+ze, Wa[32:]) )
// then atomic segment-max into emax[dst].
// ---------------------------------------------------------------------------
__global__ __launch_bounds__(256) void edge_logits(
    const float* __restrict__ zn, const float* __restrict__ ze,
    const int* __restrict__ src, const int* __restrict__ dst,
    const float* __restrict__ Wa, float* __restrict__ elog,
    float* __restrict__ emax, int E)
{
    const int edge = blockIdx.x * 8 + (threadIdx.x >> 5);
    const int lane = threadIdx.x & 31;
    if (edge >= E) return;
    const int s = src[edge], d = dst[edge];
    const float wa0 = Wa[lane], wa1 = Wa[32 + lane];
    const float* zrow = ze + (size_t)edge * HD;
    const float* srow = zn + (size_t)s * HD;
    const float* drow = zn + (size_t)d * HD;
    for (int h = 0; h < HH; ++h) {
        const int c = h * DD + lane;
        const float zev = zrow[c];
        const float vs = srow[c] + zev;
        const float vd = drow[c] + zev;
        float p = vs * wa0 + vd * wa1;
        for (int off = 16; off > 0; off >>= 1) p += __shfl_xor(p, off, 32);
        if (lane == 0) {
            float e = p > 0.f ? p : NEG_SLOPE * p;
            elog[(size_t)edge * HH + h] = e;
            atomicMaxF(&emax[d * HH + h], e);
        }
    }
}

// ---------------------------------------------------------------------------
// softmax numerator: ex = exp(e - emax[dst]); denom += ex (L2-resident atomics)
// ---------------------------------------------------------------------------
__global__ void softmax_num(const int* __restrict__ dst, float* __restrict__ elog,
                            const float* __restrict__ emax, float* __restrict__ denom,
                            int EH)
{
    int i = blockIdx.x * blockDim.x + threadIdx.x;
    if (i >= EH) return;
    const int e = i >> 2, h = i & 3;
    const int d = dst[e];
    const float ex = expf(elog[i] - emax[d * HH + h]);
    elog[i] = ex;                     // reuse buffer for ex
    atomicAdd(&denom[d * HH + h], ex);
}

// ---------------------------------------------------------------------------
// weighted scatter: agg[dst] += (ex/denom[dst]) * ze   (global_atomic_add_f32,
// 76.8M atomics into 25.6MB L2-resident buffer)
// ---------------------------------------------------------------------------
__global__ void scatter_agg(const int* __restrict__ dst, const float* __restrict__ ex,
                            const float* __restrict__ denom, const float* __restrict__ ze,
                            float* __restrict__ agg, int EHD)
{
    int i = blockIdx.x * blockDim.x + threadIdx.x;
    if (i >= EHD) return;
    const int e = i >> 7;             // / 128
    const int r = i & 127;
    const int h = r >> 5;
    const int d = dst[e];
    const float alpha = ex[e * HH + h] / denom[d * HH + h];
    atomicAdd(&agg[(size_t)d * HD + r], alpha * ze[i]);
}

// ---------------------------------------------------------------------------
// finalize: z = (deg>0 ? agg : zn); out = relu(z @ Wo)  [Wo: 128x32]
// block = 64 threads (2 waves), 16 rows per block, WMMA f32 16x16x4.
// denom > 0 <=> node has incoming edges (exp > 0 always).
// ---------------------------------------------------------------------------
__global__ __launch_bounds__(64) void out_gemm_wmma(
    const float* __restrict__ agg, const float* __restrict__ zn,
    const float* __restrict__ denom, const float* __restrict__ Wo,
    float* __restrict__ out, int N)
{
    __shared__ __align__(16) float sZ[16 * 132];
    const int tileM = blockIdx.x * 16;
    const int tid = threadIdx.x;
    for (int i = tid; i < 512; i += 64) {
        int r = i >> 5, c4 = i & 31;
        int row = tileM + r; if (row >= N) row = N - 1;
        int h = c4 >> 3;                        // head of these 4 cols
        float dg = denom[row * HH + h];
        v4f va = *reinterpret_cast<const v4f*>(agg + (size_t)row * HD + c4 * 4);
        v4f vz = *reinterpret_cast<const v4f*>(zn  + (size_t)row * HD + c4 * 4);
        *reinterpret_cast<v4f*>(&sZ[r * 132 + c4 * 4]) = (dg > 0.f) ? va : vz;
    }
    __syncthreads();

    const int wave = tid >> 5, lane = tid & 31;
    const int lo = lane & 15, hi = lane >> 4;
    const int col0 = wave << 4;                 // 0 or 16 (Ncols = 32)
    v8f acc = {0.f, 0.f, 0.f, 0.f, 0.f, 0.f, 0.f, 0.f};
    for (int k0 = 0; k0 < 128; k0 += 4) {
        const int ka = k0 + 2 * hi;
        v2f a, b;
        a.x = sZ[lo * 132 + ka];
        a.y = sZ[lo * 132 + ka + 1];
        b.x = Wo[(size_t)ka * 32 + col0 + lo];
        b.y = Wo[(size_t)(ka + 1) * 32 + col0 + lo];
        acc = __builtin_amdgcn_wmma_f32_16x16x4_f32(
            false, a, false, b, (short)0, acc, false, false);
    }
    for (int j = 0; j < 8; ++j) {
        int row = tileM + j + 8 * hi;
        if (row < N) out[(size_t)row * 32 + col0 + lo] = fmaxf(acc[j], 0.f);
    }
}

// ---------------------------------------------------------------------------
extern "C" void kernel_launch(void* const* d_in, const int* in_sizes, int n_in,
                              void* d_out, int out_size, void* d_ws, size_t ws_size,
                              hipStream_t stream)
{
    const float* node_feats = (const float*)d_in[0];   // [N,128]
    const float* edge_feats = (const float*)d_in[1];   // [E,64]
    const int*   src        = (const int*)d_in[2];     // [E]
    const int*   dst        = (const int*)d_in[3];     // [E]
    const float* Wn         = (const float*)d_in[4];   // [128,128]
    const float* We         = (const float*)d_in[5];   // [64,128]
    const float* Wa         = (const float*)d_in[6];   // [64]
    const float* Wo         = (const float*)d_in[7];   // [128,32]
    float* out = (float*)d_out;

    const int E  = in_sizes[2];
    const int IN = 128;
    const int N  = in_sizes[0] / IN;
    const int ED = in_sizes[1] / E;

    // workspace layout (floats)
    float* ws    = (float*)d_ws;
    float* zn    = ws;                 size_t o = (size_t)N * HD;   // 25.6 MB
    float* ze    = ws + o;             o += (size_t)E * HD;         // 307.2 MB
    float* elog  = ws + o;             o += (size_t)E * HH;         // 9.6 MB
    float* emax  = ws + o;             o += (size_t)N * HH;
    float* denom = ws + o;             o += (size_t)N * HH;
    float* agg   = ws + o;                                          // 25.6 MB

    // 1) projections (fp32 WMMA, TDM-staged A tiles)
    gemm_f32_wmma<<<(N + ROWS - 1) / ROWS, 256, 0, stream>>>(node_feats, Wn, zn, N, IN, HD);
    gemm_f32_wmma<<<(E + ROWS - 1) / ROWS, 256, 0, stream>>>(edge_feats, We, ze, E, ED, HD);

    // 2) init segment-softmax state
    const int NH = N * HH, NHD = N * HD;
    init_kernel<<<(NHD + 255) / 256, 256, 0, stream>>>(emax, denom, agg, NH, NHD);

    // 3) edge attention logits + segment max
    edge_logits<<<(E + 7) / 8, 256, 0, stream>>>(zn, ze, src, dst, Wa, elog, emax, E);

    // 4) softmax numerator + denominator
    const int EH = E * HH;
    softmax_num<<<(EH + 255) / 256, 256, 0, stream>>>(dst, elog, emax, denom, EH);

    // 5) weighted message scatter-sum
    const int EHD = E * HD;
    scatter_agg<<<(EHD + 255) / 256, 256, 0, stream>>>(dst, elog, denom, ze, agg, EHD);

    // 6) zero-in-degree fallback + output projection + relu (fp32 WMMA)
    out_gemm_wmma<<<(N + 15) / 16, 64, 0, stream>>>(agg, zn, denom, Wo, out, N);
}